// CenterLoss_5669356836479
// MI455X (gfx1250) — compile-verified
//
#include <hip/hip_runtime.h>

// CenterLoss on MI455X (gfx1250, wave32).
// dist[b] = ||x_b - centers[labels[b]]||^2 ; out = mean(clip(dist, 1e-12, 1e12))
// Gather-first: no B x C GEMM. HBM traffic = x (128 MB, NT-streamed);
// centers (6 MB unique) stay resident in the 192 MB L2.
// Matrix pipe: per 16-row tile, D = diff * diff^T via V_WMMA_F32_16X16X4_F32,
// diag(D) = squared distances, full f32 precision.

typedef float v2f __attribute__((ext_vector_type(2)));
typedef float v8f __attribute__((ext_vector_type(8)));

#define FEAT_DIM     2048
#define BATCH        16384
#define ROWS_PER_BLK 16
#define NWAVES       8
#define K_PER_WAVE   (FEAT_DIM / NWAVES)   // 256 -> 64 WMMA k-steps per wave

__global__ void CenterLoss_zero_out(float* out) {
    if (threadIdx.x == 0 && blockIdx.x == 0) out[0] = 0.0f;
}

__global__ __launch_bounds__(256) void CenterLoss_wmma_kernel(
    const float* __restrict__ x,          // [BATCH, FEAT_DIM]
    const long long* __restrict__ labels, // [BATCH] int64
    const float* __restrict__ centers,    // [NUM_CLASSES, FEAT_DIM]
    float* __restrict__ out)              // [1] accumulated mean
{
    __shared__ float diag_lds[NWAVES * ROWS_PER_BLK];  // 512 B

    const int tid  = threadIdx.x;
    const int lane = tid & 31;
    const int wave = tid >> 5;

    // WMMA f32 16x16x4 A-operand layout (ISA 7.12.2):
    //   lanes 0-15 : row M=lane,    VGPR0/1 = K+0, K+1
    //   lanes 16-31: row M=lane-16, VGPR0/1 = K+2, K+3
    // Symmetric operand => identical register image serves as both A and B.
    const int m     = lane & 15;            // tile row this lane feeds
    const int khalf = (lane >> 4) * 2;      // 0 for low half, 2 for high half

    const int row = blockIdx.x * ROWS_PER_BLK + m;
    const int cls = (int)labels[row];

    const float* xr = x       + (size_t)row * FEAT_DIM + khalf;
    const float* cr = centers + (size_t)cls * FEAT_DIM + khalf;
    const int    k0 = wave * K_PER_WAVE;

    v8f acc = {};

    // 64 k-steps of 4: two global_load_b64 + two v_sub_f32 + one v_wmma each.
    // x is a one-shot 128 MB stream -> non-temporal; centers hit L2.
#pragma unroll 8
    for (int k = 0; k < K_PER_WAVE; k += 4) {
        v2f xv = __builtin_nontemporal_load((const v2f*)(xr + k0 + k));
        v2f cv = *(const v2f*)(cr + k0 + k);
        v2f d;
        d.x = xv.x - cv.x;
        d.y = xv.y - cv.y;
        // D += A * B with A = B = diff tile  =>  diag(D) = row squared dists
        acc = __builtin_amdgcn_wmma_f32_16x16x4_f32(
            /*neg_a=*/false, d, /*neg_b=*/false, d,
            /*c_mod=*/(short)0, acc, /*reuse_a=*/false, /*reuse_b=*/false);
    }

    // Extract diag of the 16x16 f32 C/D tile (ISA layout):
    //   diag(r)   at VGPR r, lane r        (r = 0..7)
    //   diag(r+8) at VGPR r, lane r + 24
    float myd = 0.0f;
#pragma unroll
    for (int r = 0; r < 8; ++r) {
        float lo = __shfl(acc[r], r,      32);
        float hi = __shfl(acc[r], r + 24, 32);
        if (lane == r)     myd = lo;
        if (lane == r + 8) myd = hi;
    }
    if (lane < 16) diag_lds[wave * ROWS_PER_BLK + lane] = myd;
    __syncthreads();

    // Wave 0: sum the 8 per-wave partial diagonals, clip AFTER the full-K sum,
    // reduce the 16 row distances, and accumulate the mean contribution.
    if (wave == 0) {
        float dist = 0.0f;
        if (lane < 16) {
#pragma unroll
            for (int w = 0; w < NWAVES; ++w)
                dist += diag_lds[w * ROWS_PER_BLK + lane];
            dist = fminf(fmaxf(dist, 1e-12f), 1e12f);
        }
#pragma unroll
        for (int off = 8; off >= 1; off >>= 1)
            dist += __shfl_xor(dist, off, 32);   // lane0 = sum of lanes 0..15
        if (lane == 0)
            atomicAdd(out, dist * (1.0f / (float)BATCH));
    }
}

extern "C" void kernel_launch(void* const* d_in, const int* in_sizes, int n_in,
                              void* d_out, int out_size, void* d_ws, size_t ws_size,
                              hipStream_t stream) {
    const float*     x       = (const float*)d_in[0];
    const long long* labels  = (const long long*)d_in[1];
    const float*     centers = (const float*)d_in[2];
    float*           out     = (float*)d_out;

    CenterLoss_zero_out<<<1, 32, 0, stream>>>(out);

    dim3 grid(BATCH / ROWS_PER_BLK);   // 1024 blocks
    dim3 block(32 * NWAVES);           // 256 threads = 8 wave32s
    CenterLoss_wmma_kernel<<<grid, block, 0, stream>>>(x, labels, centers, out);
}